// GCN_14293651161340
// MI455X (gfx1250) — compile-verified
//
#include <hip/hip_runtime.h>
#include <math.h>

// ---- problem constants (from reference) ----
#define NN 100000      // nodes (== 6250 * 16, exact 16-row tiling)
#define EE 1600000     // edges
#define FIN 64
#define HH 32

typedef __attribute__((ext_vector_type(2))) float v2f;
typedef __attribute__((ext_vector_type(8))) float v8f;

// ======================================================================
// WMMA f32 GEMM:  C[N x 32] = A[N x Kin] @ W[Kin x 32]
// one wave computes one 16x16 tile of C; K stepped by 4 via
// v_wmma_f32_16x16x4_f32. N is an exact multiple of 16 so every wave is
// full (EXEC all-1s, required by WMMA).
//   A frag (16x4 f32): lanes 0-15 hold K=0,1; lanes 16-31 hold K=2,3
//   B frag (4x16 f32): lane%16 = N col; lanes<16 hold K=0,1; >=16 K=2,3
//   C/D (16x16 f32):   vgpr r holds row m = r + 8*(lane>=16), col = lane%16
// ======================================================================
__global__ void wmma_gemm_n32(const float* __restrict__ A,
                              const float* __restrict__ W,
                              float* __restrict__ C, int Kin) {
    const int wave = threadIdx.x >> 5;
    const int mt   = blockIdx.x * 4 + wave;          // 16-row tile index
    if (mt >= NN / 16) return;                       // wave-uniform exit
    const int nb   = blockIdx.y << 4;                // col base: 0 or 16
    const int lane = threadIdx.x & 31;
    const int lo   = lane & 15;
    const int hi   = lane >> 4;
    const int row  = mt * 16 + lo;

    v8f acc = {};
    for (int k0 = 0; k0 < Kin; k0 += 4) {
        const int ka = k0 + 2 * hi;
        v2f a;
        a.x = A[row * Kin + ka];
        a.y = A[row * Kin + ka + 1];
        v2f b;
        b.x = W[ka * HH + nb + lo];
        b.y = W[(ka + 1) * HH + nb + lo];
        acc = __builtin_amdgcn_wmma_f32_16x16x4_f32(
            /*neg_a=*/false, a, /*neg_b=*/false, b,
            /*c_mod=*/(short)0, acc, /*reuse_a=*/false, /*reuse_b=*/false);
    }
#pragma unroll
    for (int r = 0; r < 8; ++r) {
        const int m = mt * 16 + r + 8 * hi;
        C[m * HH + nb + lo] = acc[r];
    }
}

// ---- edge scatter: agg[dst] += ew * feat[src], d = 32 (lane per feature) ----
__global__ void edge_scatter32(const float* __restrict__ feat,
                               const int* __restrict__ src,
                               const int* __restrict__ dst,
                               const float* __restrict__ ew,
                               float* __restrict__ agg) {
    const long long tid = (long long)blockIdx.x * blockDim.x + threadIdx.x;
    const int e = (int)(tid >> 5);
    const int f = (int)(tid & 31);
    if (e >= EE) return;
    const float v = feat[src[e] * HH + f] * ew[e];
    atomicAdd(&agg[dst[e] * HH + f], v);
}

// ---- edge scatter, d = 2 (fused policy/value channels) ----
__global__ void edge_scatter2(const float* __restrict__ feat,
                              const int* __restrict__ src,
                              const int* __restrict__ dst,
                              const float* __restrict__ ew,
                              float* __restrict__ agg) {
    const long long tid = (long long)blockIdx.x * blockDim.x + threadIdx.x;
    const int e = (int)(tid >> 1);
    const int c = (int)(tid & 1);
    if (e >= EE) return;
    const float v = feat[src[e] * 2 + c] * ew[e];
    atomicAdd(&agg[dst[e] * 2 + c], v);
}

// ---- h = relu(agg + root + bias) over N*32 ----
__global__ void relu_combine(const float* __restrict__ agg,
                             const float* __restrict__ root,
                             const float* __restrict__ bias,
                             float* __restrict__ h) {
    const int i = blockIdx.x * blockDim.x + threadIdx.x;
    if (i >= NN * HH) return;
    const float v = agg[i] + root[i] + bias[i & (HH - 1)];
    h[i] = v > 0.0f ? v : 0.0f;
}

// ---- per-node 32->2 projections for policy/value (rel and root) ----
__global__ void pv_transform(const float* __restrict__ h,
                             const float* __restrict__ Wp_rel,
                             const float* __restrict__ Wv_rel,
                             const float* __restrict__ Wp_root,
                             const float* __restrict__ Wv_root,
                             float* __restrict__ pvrel,
                             float* __restrict__ pvroot) {
    const int i = blockIdx.x * blockDim.x + threadIdx.x;
    if (i >= NN) return;
    const float* hv = h + i * HH;
    float pr = 0.f, vr = 0.f, po = 0.f, vo = 0.f;
#pragma unroll
    for (int k = 0; k < HH; ++k) {
        const float hk = hv[k];
        pr += hk * Wp_rel[k];
        vr += hk * Wv_rel[k];
        po += hk * Wp_root[k];
        vo += hk * Wv_root[k];
    }
    pvrel[i * 2 + 0] = pr;
    pvrel[i * 2 + 1] = vr;
    pvroot[i * 2 + 0] = po;
    pvroot[i * 2 + 1] = vo;
}

// ---- p,v = agg + root + bias ----
__global__ void finalize_pv(const float* __restrict__ pvagg,
                            const float* __restrict__ pvroot,
                            const float* __restrict__ bp,
                            const float* __restrict__ bv,
                            float* __restrict__ pbuf,
                            float* __restrict__ vbuf) {
    const int i = blockIdx.x * blockDim.x + threadIdx.x;
    if (i >= NN) return;
    pbuf[i] = pvagg[i * 2 + 0] + pvroot[i * 2 + 0] + bp[0];
    vbuf[i] = pvagg[i * 2 + 1] + pvroot[i * 2 + 1] + bv[0];
}

// ---- legal-move mask: nodes reachable from current_node ----
__global__ void build_mask(const int* __restrict__ src,
                           const int* __restrict__ dst,
                           const int* __restrict__ cur,
                           int* __restrict__ mask) {
    const int e = blockIdx.x * blockDim.x + threadIdx.x;
    if (e >= EE) return;
    if (src[e] == cur[0]) mask[dst[e]] = 1;   // racing stores of identical value
}

// ---- reduction stage 1: per-block max of masked p (p_m != 0 entries) ----
__global__ void reduce_max_stage1(const float* __restrict__ p,
                                  const int* __restrict__ mask,
                                  float* __restrict__ partials) {
    __shared__ float s[256];
    float m = -INFINITY;
    for (int i = blockIdx.x * blockDim.x + threadIdx.x; i < NN;
         i += gridDim.x * blockDim.x) {
        const float pm = mask[i] ? p[i] : 0.0f;
        if (pm != 0.0f) m = fmaxf(m, pm);
    }
    s[threadIdx.x] = m;
    __syncthreads();
    for (int st = 128; st > 0; st >>= 1) {
        if (threadIdx.x < st) s[threadIdx.x] = fmaxf(s[threadIdx.x], s[threadIdx.x + st]);
        __syncthreads();
    }
    if (threadIdx.x == 0) partials[blockIdx.x] = s[0];
}

__global__ void reduce_max_stage2(const float* __restrict__ partials,
                                  float* __restrict__ scal) {
    __shared__ float s[256];
    float m = -INFINITY;
    for (int i = threadIdx.x; i < 1024; i += 256) m = fmaxf(m, partials[i]);
    s[threadIdx.x] = m;
    __syncthreads();
    for (int st = 128; st > 0; st >>= 1) {
        if (threadIdx.x < st) s[threadIdx.x] = fmaxf(s[threadIdx.x], s[threadIdx.x + st]);
        __syncthreads();
    }
    if (threadIdx.x == 0) scal[0] = s[0];
}

// ---- reduction: sum of exp(p_m - gmax) over included entries ----
__global__ void reduce_sum_stage1(const float* __restrict__ p,
                                  const int* __restrict__ mask,
                                  const float* __restrict__ scal,
                                  float* __restrict__ partials) {
    __shared__ float s[256];
    const float gmax = scal[0];
    float acc = 0.0f;
    for (int i = blockIdx.x * blockDim.x + threadIdx.x; i < NN;
         i += gridDim.x * blockDim.x) {
        const float pm = mask[i] ? p[i] : 0.0f;
        if (pm != 0.0f) acc += __expf(pm - gmax);
    }
    s[threadIdx.x] = acc;
    __syncthreads();
    for (int st = 128; st > 0; st >>= 1) {
        if (threadIdx.x < st) s[threadIdx.x] += s[threadIdx.x + st];
        __syncthreads();
    }
    if (threadIdx.x == 0) partials[blockIdx.x] = s[0];
}

__global__ void reduce_sum_stage2(const float* __restrict__ partials,
                                  float* __restrict__ scal) {
    __shared__ float s[256];
    float acc = 0.0f;
    for (int i = threadIdx.x; i < 1024; i += 256) acc += partials[i];
    s[threadIdx.x] = acc;
    __syncthreads();
    for (int st = 128; st > 0; st >>= 1) {
        if (threadIdx.x < st) s[threadIdx.x] += s[threadIdx.x + st];
        __syncthreads();
    }
    if (threadIdx.x == 0) scal[1] = s[0];
}

// ---- final outputs: softmax(p_masked) and mask*v ----
__global__ void write_out(const float* __restrict__ p,
                          const float* __restrict__ v,
                          const int* __restrict__ mask,
                          const float* __restrict__ scal,
                          float* __restrict__ out) {
    const int i = blockIdx.x * blockDim.x + threadIdx.x;
    if (i >= NN) return;
    const float gmax = scal[0];
    const float gsum = scal[1];
    const float pm = mask[i] ? p[i] : 0.0f;
    out[i] = (pm != 0.0f) ? __expf(pm - gmax) / gsum : 0.0f;
    out[NN + i] = mask[i] ? v[i] : 0.0f;
}

extern "C" void kernel_launch(void* const* d_in, const int* in_sizes, int n_in,
                              void* d_out, int out_size, void* d_ws, size_t ws_size,
                              hipStream_t stream) {
    const float* x        = (const float*)d_in[0];
    const int*   ei       = (const int*)d_in[1];     // [2,E] int32
    const float* ea       = (const float*)d_in[2];
    const int*   cur      = (const int*)d_in[3];
    const float* Win_rel  = (const float*)d_in[4];
    const float* bin_rel  = (const float*)d_in[5];
    const float* Win_root = (const float*)d_in[6];
    const float* Wh_rel   = (const float*)d_in[7];
    const float* bh_rel   = (const float*)d_in[8];
    const float* Wh_root  = (const float*)d_in[9];
    const float* Wp_rel   = (const float*)d_in[10];
    const float* bp       = (const float*)d_in[11];
    const float* Wp_root  = (const float*)d_in[12];
    const float* Wv_rel   = (const float*)d_in[13];
    const float* bv       = (const float*)d_in[14];
    const float* Wv_root  = (const float*)d_in[15];
    const int* src = ei;
    const int* dst = ei + EE;
    float* out = (float*)d_out;

    // workspace carve-up (floats)
    float* ws       = (float*)d_ws;
    float* frel     = ws;                       // N*32
    float* froot    = frel   + (size_t)NN * HH; // N*32
    float* agg      = froot  + (size_t)NN * HH; // N*32
    float* h        = agg    + (size_t)NN * HH; // N*32
    float* pvrel    = h      + (size_t)NN * HH; // N*2
    float* pvroot   = pvrel  + (size_t)NN * 2;  // N*2
    float* pvagg    = pvroot + (size_t)NN * 2;  // N*2
    float* pbuf     = pvagg  + (size_t)NN * 2;  // N
    float* vbuf     = pbuf   + (size_t)NN;      // N
    int*   mask     = (int*)(vbuf + (size_t)NN);// N
    float* partials = (float*)(mask + (size_t)NN); // 1024
    float* scal     = partials + 1024;          // 2

    const dim3 gemmGrid((NN / 16 + 3) / 4, 2);  // 6250 row tiles / 4 waves, 2 col tiles
    const dim3 gemmBlk(128);
    const int sc32Blocks = (int)(((long long)EE * 32 + 255) / 256);
    const int sc2Blocks  = (int)(((long long)EE * 2 + 255) / 256);
    const int nhBlocks   = (NN * HH + 255) / 256;
    const int nBlocks    = (NN + 255) / 256;
    const int eBlocks    = (EE + 255) / 256;

    // ---- layer 1: GraphConv(64 -> 32), W_rel pushed before aggregation ----
    wmma_gemm_n32<<<gemmGrid, gemmBlk, 0, stream>>>(x, Win_rel,  frel,  FIN);
    wmma_gemm_n32<<<gemmGrid, gemmBlk, 0, stream>>>(x, Win_root, froot, FIN);
    hipMemsetAsync(agg, 0, (size_t)NN * HH * sizeof(float), stream);
    edge_scatter32<<<sc32Blocks, 256, 0, stream>>>(frel, src, dst, ea, agg);
    relu_combine<<<nhBlocks, 256, 0, stream>>>(agg, froot, bin_rel, h);

    // ---- layer 2: GraphConv(32 -> 32) ----
    wmma_gemm_n32<<<gemmGrid, gemmBlk, 0, stream>>>(h, Wh_rel,  frel,  HH);
    wmma_gemm_n32<<<gemmGrid, gemmBlk, 0, stream>>>(h, Wh_root, froot, HH);
    hipMemsetAsync(agg, 0, (size_t)NN * HH * sizeof(float), stream);
    edge_scatter32<<<sc32Blocks, 256, 0, stream>>>(frel, src, dst, ea, agg);
    relu_combine<<<nhBlocks, 256, 0, stream>>>(agg, froot, bh_rel, h);

    // ---- policy/value convs (32 -> 1 each), fused into 2 channels ----
    pv_transform<<<nBlocks, 256, 0, stream>>>(h, Wp_rel, Wv_rel, Wp_root, Wv_root,
                                              pvrel, pvroot);
    hipMemsetAsync(pvagg, 0, (size_t)NN * 2 * sizeof(float), stream);
    edge_scatter2<<<sc2Blocks, 256, 0, stream>>>(pvrel, src, dst, ea, pvagg);
    finalize_pv<<<nBlocks, 256, 0, stream>>>(pvagg, pvroot, bp, bv, pbuf, vbuf);

    // ---- legal-move mask ----
    hipMemsetAsync(mask, 0, (size_t)NN * sizeof(int), stream);
    build_mask<<<eBlocks, 256, 0, stream>>>(src, dst, cur, mask);

    // ---- masked softmax over p, masked v ----
    reduce_max_stage1<<<1024, 256, 0, stream>>>(pbuf, mask, partials);
    reduce_max_stage2<<<1, 256, 0, stream>>>(partials, scal);
    reduce_sum_stage1<<<1024, 256, 0, stream>>>(pbuf, mask, scal, partials);
    reduce_sum_stage2<<<1, 256, 0, stream>>>(partials, scal);
    write_out<<<nBlocks, 256, 0, stream>>>(pbuf, vbuf, mask, scal, out);
}